// cross_entropy_with_hnm_for_one_class_detection_55825984913685
// MI455X (gfx1250) — compile-verified
//
#include <hip/hip_runtime.h>
#include <hip/hip_bf16.h>

typedef __attribute__((ext_vector_type(2))) float v2f;
typedef __attribute__((ext_vector_type(8))) float v8f;

// Full 32-lane (wave32) sum using V_WMMA_F32_16X16X4_F32 with an all-ones B.
// A layout (16x4 f32): lanes 0-15 VGPR0 -> K=0, VGPR1 -> K=1; lanes 16-31
// VGPR0 -> K=2, VGPR1 -> K=3, M = lane%16.  With a = {val, 0}, b = {1,1}:
//   D[m][n] = val_m + val_{m+16}  (independent of n)
// C/D layout: lane L<16 VGPR v = D[v][L]; lane L>=16 VGPR v = D[v+8][L-16].
// Sum of 8 C regs gives half-sum; xor-16 shuffle adds the complementary half.
__device__ __forceinline__ float wave_sum_wmma(float val) {
  v2f a; a[0] = val;  a[1] = 0.0f;
  v2f b; b[0] = 1.0f; b[1] = 1.0f;
  v8f c = {};
  c = __builtin_amdgcn_wmma_f32_16x16x4_f32(false, a, false, b, (short)0, c,
                                            false, false);
  float s = ((c[0] + c[1]) + (c[2] + c[3])) + ((c[4] + c[5]) + (c[6] + c[7]));
  s += __shfl_xor(s, 16, 32);
  return s;  // every lane holds the 32-lane total
}

__device__ __forceinline__ void ld4(const float* __restrict__ p, float o[4]) {
  float4 t = *reinterpret_cast<const float4*>(p);  // global_load_b128
  o[0] = t.x; o[1] = t.y; o[2] = t.z; o[3] = t.w;
}

__device__ __forceinline__ void st4(float* __restrict__ p, const float o[4]) {
  float4 t; t.x = o[0]; t.y = o[1]; t.z = o[2]; t.w = o[3];
  *reinterpret_cast<float4*>(p) = t;               // global_store_b128
}

__global__ void zero_u32_kernel(unsigned* __restrict__ p, int n) {
  int i = blockIdx.x * blockDim.x + threadIdx.x;
  if (i < n) p[i] = 0u;
}

// Streaming pass (vectorized x4): softmax / log-softmax, neg_prob & neg_ce
// scratch, pos/neg counts, pos-CE sum, bbox mse & mask sums, + radix pass-1
// histogram (neg_prob bits [31:21], 2048 bins).  HW and N are multiples of 4,
// so a 4-wide chunk never crosses a batch plane and is 16B aligned.
__global__ __launch_bounds__(256)
void prep_kernel(const float* __restrict__ score, const float* __restrict__ bbox,
                 const float* __restrict__ gmask, const float* __restrict__ glabel,
                 float* __restrict__ neg_prob, float* __restrict__ neg_ce,
                 float* __restrict__ partials, unsigned* __restrict__ hist1,
                 int HW, int N) {
  __shared__ unsigned lhist[2048];
  __shared__ float lred[8 * 5];
  for (int i = threadIdx.x; i < 2048; i += 256) lhist[i] = 0u;
  __syncthreads();

  float acc_pos = 0.f, acc_neg = 0.f, acc_ce = 0.f, acc_mse = 0.f, acc_msk = 0.f;
  const int stride = gridDim.x * 256 * 4;
  for (int i = (blockIdx.x * 256 + threadIdx.x) * 4; i < N; i += stride) {
    int bb = i / HW;
    int hw = i - bb * HW;
    const float* sc = score + (size_t)bb * 2 * HW + hw;
    const float* gl = glabel + (size_t)bb * 6 * HW + hw;
    const float* gm = gmask + (size_t)bb * 6 * HW + hw + 2 * HW;
    const float* pb = bbox + (size_t)bb * 4 * HW + hw;

    float s0[4], s1[4], l0[4], l1[4], npv[4], ncv[4];
    ld4(sc, s0);
    ld4(sc + HW, s1);
    ld4(gl, l0);
    ld4(gl + HW, l1);
#pragma unroll
    for (int j = 0; j < 4; ++j) {
      float m = fmaxf(s0[j], s1[j]);
      float e0 = expf(s0[j] - m), e1 = expf(s1[j] - m);
      float den = e0 + e1;
      float logden = logf(den);
      float log_sm0 = (s0[j] - m) - logden;
      float log_sm1 = (s1[j] - m) - logden;
      float sm1 = e1 / den;
      bool pos = l0[j] > 0.5f;
      bool neg = l1[j] > 0.5f;
      float np = neg ? sm1 : 0.0f;
      npv[j] = np;
      ncv[j] = neg ? (-l1[j] * log_sm1) : 0.0f;
      acc_pos += pos ? 1.0f : 0.0f;
      acc_neg += neg ? 1.0f : 0.0f;
      acc_ce += pos ? (-l0[j] * log_sm0) : 0.0f;
      atomicAdd(&lhist[__float_as_uint(np) >> 21], 1u);
    }
    st4(neg_prob + i, npv);
    st4(neg_ce + i, ncv);

#pragma unroll
    for (int c = 0; c < 4; ++c) {
      float mk[4], gt[4], pv[4];
      ld4(gm + c * HW, mk);
      ld4(gl + (2 + c) * HW, gt);
      ld4(pb + c * HW, pv);
#pragma unroll
      for (int j = 0; j < 4; ++j) {
        float d = gt[j] * mk[j] - pv[j] * mk[j];
        acc_mse += d * d;
        acc_msk += mk[j];
      }
    }
  }
  __syncthreads();
  for (int i = threadIdx.x; i < 2048; i += 256) {
    unsigned v = lhist[i];
    if (v) atomicAdd(&hist1[i], v);
  }

  // block reduction of the 5 accumulators (all 256 threads converged here)
  float vals[5] = {acc_pos, acc_neg, acc_ce, acc_mse, acc_msk};
  int wid = threadIdx.x >> 5;
  int lane = threadIdx.x & 31;
#pragma unroll
  for (int j = 0; j < 5; ++j) {
    float s = wave_sum_wmma(vals[j]);
    if (lane == 0) lred[wid * 5 + j] = s;
  }
  __syncthreads();
  if (threadIdx.x < 5) {
    float t = 0.f;
#pragma unroll
    for (int wv = 0; wv < 8; ++wv) t += lred[wv * 5 + threadIdx.x];
    partials[(size_t)blockIdx.x * 5 + threadIdx.x] = t;
  }
}

// Single block: reduce block partials (fixed order -> deterministic), compute
// k = min(10*pos, neg) as int32 (truncation, clamped), scan pass-1 histogram,
// and zero hist2 for the next pass.
__global__ __launch_bounds__(256)
void scanA_kernel(const float* __restrict__ partials, int G,
                  const unsigned* __restrict__ hist1, unsigned* __restrict__ hist2,
                  float* __restrict__ stats, int N) {
  __shared__ float lds[256];
  __shared__ float comp[5];
  for (int j = 0; j < 5; ++j) {
    float a = 0.f;
    for (int b = threadIdx.x; b < G; b += 256) a += partials[(size_t)b * 5 + j];
    lds[threadIdx.x] = a;
    __syncthreads();
    for (int s = 128; s > 0; s >>= 1) {
      if (threadIdx.x < s) lds[threadIdx.x] += lds[threadIdx.x + s];
      __syncthreads();
    }
    if (threadIdx.x == 0) comp[j] = lds[0];
    __syncthreads();
  }
  for (int i = threadIdx.x; i < 2048; i += 256) hist2[i] = 0u;
  if (threadIdx.x == 0) {
    stats[0] = comp[0];
    stats[1] = comp[1];
    stats[2] = comp[2];
    stats[3] = comp[3];
    stats[4] = comp[4];
    float nsf = fminf(10.0f * comp[0], comp[1]);
    int k = (int)nsf;  // truncation, matches astype(int32)
    if (k < 0) k = 0;
    if (k > N - 1) k = N - 1;
    unsigned kk = (unsigned)k, cum = 0, d = 0;
    for (d = 0; d < 2048; ++d) {
      unsigned c = hist1[d];
      if (cum + c > kk) break;
      cum += c;
    }
    unsigned* st = (unsigned*)stats;
    st[5] = d << 21;   // prefix
    st[6] = kk - cum;  // remaining rank
  }
}

__global__ __launch_bounds__(256)
void hist_mid_kernel(const float* __restrict__ neg_prob, int N,
                     const float* __restrict__ stats, unsigned* __restrict__ hist2) {
  __shared__ unsigned lhist[2048];
  unsigned top = ((const unsigned*)stats)[5] >> 21;
  for (int i = threadIdx.x; i < 2048; i += 256) lhist[i] = 0u;
  __syncthreads();
  const int stride = gridDim.x * 256 * 4;
  for (int i = (blockIdx.x * 256 + threadIdx.x) * 4; i < N; i += stride) {
    float np[4];
    ld4(neg_prob + i, np);
#pragma unroll
    for (int j = 0; j < 4; ++j) {
      unsigned key = __float_as_uint(np[j]);
      if ((key >> 21) == top) atomicAdd(&lhist[(key >> 10) & 2047u], 1u);
    }
  }
  __syncthreads();
  for (int i = threadIdx.x; i < 2048; i += 256) {
    unsigned v = lhist[i];
    if (v) atomicAdd(&hist2[i], v);
  }
}

__global__ __launch_bounds__(256)
void scanB_kernel(const unsigned* __restrict__ hist2, unsigned* __restrict__ hist3,
                  float* __restrict__ stats) {
  for (int i = threadIdx.x; i < 1024; i += 256) hist3[i] = 0u;
  if (threadIdx.x == 0) {
    unsigned* st = (unsigned*)stats;
    unsigned kk = st[6], cum = 0, d = 0;
    for (d = 0; d < 2048; ++d) {
      unsigned c = hist2[d];
      if (cum + c > kk) break;
      cum += c;
    }
    st[5] |= d << 10;
    st[6] = kk - cum;
  }
}

__global__ __launch_bounds__(256)
void hist_low_kernel(const float* __restrict__ neg_prob, int N,
                     const float* __restrict__ stats, unsigned* __restrict__ hist3) {
  __shared__ unsigned lhist[1024];
  unsigned top = ((const unsigned*)stats)[5] >> 10;
  for (int i = threadIdx.x; i < 1024; i += 256) lhist[i] = 0u;
  __syncthreads();
  const int stride = gridDim.x * 256 * 4;
  for (int i = (blockIdx.x * 256 + threadIdx.x) * 4; i < N; i += stride) {
    float np[4];
    ld4(neg_prob + i, np);
#pragma unroll
    for (int j = 0; j < 4; ++j) {
      unsigned key = __float_as_uint(np[j]);
      if ((key >> 10) == top) atomicAdd(&lhist[key & 1023u], 1u);
    }
  }
  __syncthreads();
  for (int i = threadIdx.x; i < 1024; i += 256) {
    unsigned v = lhist[i];
    if (v) atomicAdd(&hist3[i], v);
  }
}

__global__ void scanC_kernel(const unsigned* __restrict__ hist3,
                             float* __restrict__ stats) {
  if (threadIdx.x == 0 && blockIdx.x == 0) {
    unsigned* st = (unsigned*)stats;
    unsigned kk = st[6], cum = 0, d = 0;
    for (d = 0; d < 1024; ++d) {
      unsigned c = hist3[d];
      if (cum + c > kk) break;
      cum += c;
    }
    st[7] = st[5] | d;  // exact bits of the k-th smallest neg_prob
  }
}

__global__ __launch_bounds__(256)
void negce_kernel(const float* __restrict__ neg_prob, const float* __restrict__ neg_ce,
                  int N, const float* __restrict__ stats,
                  float* __restrict__ negpart) {
  float thr = __uint_as_float(((const unsigned*)stats)[7]);
  float acc = 0.f;
  const int stride = gridDim.x * 256 * 4;
  for (int i = (blockIdx.x * 256 + threadIdx.x) * 4; i < N; i += stride) {
    float np[4], nc[4];
    ld4(neg_prob + i, np);
    ld4(neg_ce + i, nc);
#pragma unroll
    for (int j = 0; j < 4; ++j) acc += (np[j] <= thr) ? nc[j] : 0.0f;
  }
  __shared__ float lred[8];
  float s = wave_sum_wmma(acc);  // all 256 threads converged, EXEC all ones
  int wid = threadIdx.x >> 5, lane = threadIdx.x & 31;
  if (lane == 0) lred[wid] = s;
  __syncthreads();
  if (threadIdx.x == 0) {
    float t = 0.f;
#pragma unroll
    for (int i = 0; i < 8; ++i) t += lred[i];
    negpart[blockIdx.x] = t;
  }
}

__global__ __launch_bounds__(256)
void final_kernel(const float* __restrict__ negpart, int G,
                  const float* __restrict__ stats, float* __restrict__ out,
                  int scale_idx, int N, float w_dim) {
  __shared__ float lds[256];
  float a = 0.f;
  for (int b = threadIdx.x; b < G; b += 256) a += negpart[b];
  lds[threadIdx.x] = a;
  __syncthreads();
  for (int s = 128; s > 0; s >>= 1) {
    if (threadIdx.x < s) lds[threadIdx.x] += lds[threadIdx.x + s];
    __syncthreads();
  }
  if (threadIdx.x == 0) {
    float ce_sum = stats[2] + lds[0];
    float loss_score = ce_sum / (2.0f * (float)N);  // ce.size = b*2*h*w
    float mse_sum = stats[3] / w_dim;               // mean over axis=-1 (w)
    float loss_bbox = mse_sum / stats[4];
    out[scale_idx] = loss_score + loss_bbox;
  }
}

extern "C" void kernel_launch(void* const* d_in, const int* in_sizes, int n_in,
                              void* d_out, int out_size, void* d_ws, size_t ws_size,
                              hipStream_t stream) {
  (void)in_sizes; (void)n_in; (void)out_size; (void)ws_size;
  const int HW[3] = {160 * 160, 80 * 80, 40 * 40};
  const float WD[3] = {160.f, 80.f, 40.f};

  char* w = (char*)d_ws;
  size_t off = 0;
  auto alloc = [&](size_t bytes) {
    size_t o = off;
    off = (off + bytes + 255) & ~(size_t)255;
    return o;
  };
  size_t stats_off = alloc(3 * 64);
  size_t h1_off = alloc(3 * 2048 * sizeof(unsigned));
  size_t h2_off = alloc(3 * 2048 * sizeof(unsigned));
  size_t h3_off = alloc(3 * 1024 * sizeof(unsigned));
  size_t parts_off = alloc(3 * 1024 * 5 * sizeof(float));
  size_t negp_off = alloc(3 * 1024 * sizeof(float));
  size_t np_off[3], nc_off[3];
  for (int s = 0; s < 3; ++s) {
    size_t N = (size_t)32 * HW[s];
    np_off[s] = alloc(N * sizeof(float));
    nc_off[s] = alloc(N * sizeof(float));
  }

  // zero pass-1 histograms for all 3 scales (hist2/hist3 are zeroed by the
  // scan kernels of the same scale; every other scratch word is overwritten)
  zero_u32_kernel<<<(3 * 2048 + 255) / 256, 256, 0, stream>>>(
      (unsigned*)(w + h1_off), 3 * 2048);

  for (int s = 0; s < 3; ++s) {
    int N = 32 * HW[s];
    int G = (N + 2047) / 2048;  // 8 elements (2 float4 iters) per thread
    if (G > 1024) G = 1024;
    const float* score = (const float*)d_in[4 * s + 0];
    const float* bbox = (const float*)d_in[4 * s + 1];
    const float* gmask = (const float*)d_in[4 * s + 2];
    const float* glabel = (const float*)d_in[4 * s + 3];
    float* stats = (float*)(w + stats_off + (size_t)s * 64);
    unsigned* hist1 = (unsigned*)(w + h1_off) + (size_t)s * 2048;
    unsigned* hist2 = (unsigned*)(w + h2_off) + (size_t)s * 2048;
    unsigned* hist3 = (unsigned*)(w + h3_off) + (size_t)s * 1024;
    float* parts = (float*)(w + parts_off) + (size_t)s * 1024 * 5;
    float* negpart = (float*)(w + negp_off) + (size_t)s * 1024;
    float* nprob = (float*)(w + np_off[s]);
    float* nce = (float*)(w + nc_off[s]);

    prep_kernel<<<G, 256, 0, stream>>>(score, bbox, gmask, glabel, nprob, nce,
                                       parts, hist1, HW[s], N);
    scanA_kernel<<<1, 256, 0, stream>>>(parts, G, hist1, hist2, stats, N);
    hist_mid_kernel<<<G, 256, 0, stream>>>(nprob, N, stats, hist2);
    scanB_kernel<<<1, 256, 0, stream>>>(hist2, hist3, stats);
    hist_low_kernel<<<G, 256, 0, stream>>>(nprob, N, stats, hist3);
    scanC_kernel<<<1, 32, 0, stream>>>(hist3, stats);
    negce_kernel<<<G, 256, 0, stream>>>(nprob, nce, N, stats, negpart);
    final_kernel<<<1, 256, 0, stream>>>(negpart, G, stats, (float*)d_out, s, N,
                                        WD[s]);
  }
}